// GNN_40355512713284
// MI455X (gfx1250) — compile-verified
//
#include <hip/hip_runtime.h>
#include <hip/hip_bf16.h>
#include <math.h>

// ---------------------------------------------------------------------------
// GCN (2x GCNConv + linear + log_softmax) for MI455X / gfx1250, wave32.
//   N=100000 nodes, E=1600000 edges, IN=128, HID=64, OUT=10.
// GEMMs use V_WMMA_F32_16X16X4_F32 (fp32 in/out -> matches fp32 reference).
// Scatter-add is fp32 global atomics; working set (~51MB) is L2-resident.
// Workspace usage: dis (400KB) + xw (25.6MB) + agg/h (25.6MB) ~= 52MB.
// ---------------------------------------------------------------------------

#define N_NODES 100000
#define N_EDGES 1600000
#define IN_DIM  128
#define HID     64
#define OUT_DIM 10

typedef __attribute__((ext_vector_type(2))) float v2f;
typedef __attribute__((ext_vector_type(8))) float v8f;

// ---------------- degree / normalization ----------------
__global__ void deg_init_kernel(float* __restrict__ deg, int n) {
    int i = blockIdx.x * blockDim.x + threadIdx.x;
    if (i < n) deg[i] = 1.0f;                       // +1 for self loop
}

__global__ void deg_accum_kernel(const long long* __restrict__ col,
                                 float* __restrict__ deg, int e) {
    int i = blockIdx.x * blockDim.x + threadIdx.x;
    if (i < e) atomicAdd(&deg[(int)col[i]], 1.0f);
}

__global__ void rsqrt_kernel(float* __restrict__ deg, int n) {
    int i = blockIdx.x * blockDim.x + threadIdx.x;
    if (i < n) deg[i] = rsqrtf(deg[i]);             // in place: deg -> dis
}

// ---------------- WMMA GEMM: C[M x Nh] = A[M x K] @ W[K x Nh] --------------
// 256 threads = 8 waves; wave (mt,nt) computes the 16x16 tile at
// (blockIdx.x*32 + mt*16, nt*16).  M must be a multiple of 32 (100000 is).
template <int K, int Nh>
__global__ __launch_bounds__(256)
void gemm_wmma_kernel(const float* __restrict__ A, const float* __restrict__ W,
                      float* __restrict__ C) {
    __shared__ float sW[K * Nh];
    const int tid = threadIdx.x;
    for (int i = tid; i < K * Nh; i += 256) sW[i] = W[i];
    __syncthreads();

    const int wave = tid >> 5;
    const int lane = tid & 31;
    const int half = lane >> 4;      // 0: lanes 0-15, 1: lanes 16-31
    const int l16  = lane & 15;
    const int Mbase = blockIdx.x * 32 + (wave >> 2) * 16;
    const int Nbase = (wave & 3) * 16;

    // A layout (16x4 fp32): lane m=l16, VGPR j = A[m][k0 + 2*half + j]
    const float* arow = A + (size_t)(Mbase + l16) * K + 2 * half;

    v8f acc = {0.f, 0.f, 0.f, 0.f, 0.f, 0.f, 0.f, 0.f};
    #pragma unroll 4
    for (int k0 = 0; k0 < K; k0 += 4) {
        v2f a = *(const v2f*)(arow + k0);           // 8B aligned global load
        const int kb = k0 + 2 * half;
        v2f b;
        b.x = sW[kb * Nh + Nbase + l16];            // B: lane n=l16, K striped
        b.y = sW[(kb + 1) * Nh + Nbase + l16];
        acc = __builtin_amdgcn_wmma_f32_16x16x4_f32(
            /*neg_a=*/false, a, /*neg_b=*/false, b,
            /*c_mod=*/(short)0, acc, /*reuse_a=*/false, /*reuse_b=*/false);
    }

    // C/D layout: VGPR v -> row Mbase + v + 8*half, col Nbase + l16
    float* crow = C + (size_t)Mbase * Nh + Nbase + l16;
    #pragma unroll
    for (int v = 0; v < 8; ++v)
        crow[(size_t)(v + 8 * half) * Nh] = acc[v];
}

// ---------------- self-loop init: agg = (dis^2) * xw -----------------------
__global__ void self_loop_kernel(const float* __restrict__ dis,
                                 const float* __restrict__ xw,
                                 float* __restrict__ agg, int n) {
    int t = blockIdx.x * blockDim.x + threadIdx.x;
    int i = t >> 4, hq = (t & 15) << 2;
    if (i >= n) return;
    float d = dis[i];
    float w = d * d;
    const float4 v = *(const float4*)(xw + (size_t)i * HID + hq);
    float4 o = make_float4(w * v.x, w * v.y, w * v.z, w * v.w);
    *(float4*)(agg + (size_t)i * HID + hq) = o;
}

// ---------------- edge scatter: agg[col] += dis[row]*dis[col]*xw[row] ------
// 16 threads per edge, float4 per thread; gathers are 256B contiguous per
// half-wave and resolve in the 192MB L2 (xw is 25.6MB).
__global__ void edge_scatter_kernel(const long long* __restrict__ row,
                                    const long long* __restrict__ col,
                                    const float* __restrict__ dis,
                                    const float* __restrict__ xw,
                                    float* __restrict__ agg, int e) {
    int t = blockIdx.x * blockDim.x + threadIdx.x;
    int ei = t >> 4, hq = (t & 15) << 2;
    if (ei >= e) return;
    int r = (int)row[ei];
    int c = (int)col[ei];
    float w = dis[r] * dis[c];
    const float4 v = *(const float4*)(xw + (size_t)r * HID + hq);
    float* dst = agg + (size_t)c * HID + hq;
    atomicAdd(dst + 0, w * v.x);
    atomicAdd(dst + 1, w * v.y);
    atomicAdd(dst + 2, w * v.z);
    atomicAdd(dst + 3, w * v.w);
}

// ---------------- bias + relu (in place) -----------------------------------
__global__ void bias_relu_kernel(float* __restrict__ h,
                                 const float* __restrict__ b, int n) {
    int t = blockIdx.x * blockDim.x + threadIdx.x;
    int i = t >> 4, hq = (t & 15) << 2;
    if (i >= n) return;
    float4 v = *(float4*)(h + (size_t)i * HID + hq);
    v.x = fmaxf(v.x + b[hq + 0], 0.f);
    v.y = fmaxf(v.y + b[hq + 1], 0.f);
    v.z = fmaxf(v.z + b[hq + 2], 0.f);
    v.w = fmaxf(v.w + b[hq + 3], 0.f);
    *(float4*)(h + (size_t)i * HID + hq) = v;
}

// ---------------- logits + log_softmax -------------------------------------
__global__ __launch_bounds__(256)
void logits_lsm_kernel(const float* __restrict__ h, const float* __restrict__ Wl,
                       const float* __restrict__ bl, float* __restrict__ out,
                       int n) {
    __shared__ float sW[HID * OUT_DIM];
    __shared__ float sb[OUT_DIM];
    const int tid = threadIdx.x;
    for (int i = tid; i < HID * OUT_DIM; i += 256) sW[i] = Wl[i];
    if (tid < OUT_DIM) sb[tid] = bl[tid];
    __syncthreads();

    int node = blockIdx.x * 256 + tid;
    if (node >= n) return;
    const float* hr = h + (size_t)node * HID;

    float acc[OUT_DIM];
    #pragma unroll
    for (int j = 0; j < OUT_DIM; ++j) acc[j] = sb[j];

    #pragma unroll 4
    for (int k4 = 0; k4 < HID; k4 += 4) {
        float4 hv = *(const float4*)(hr + k4);
        const float* w0 = &sW[k4 * OUT_DIM];
        #pragma unroll
        for (int j = 0; j < OUT_DIM; ++j) {
            acc[j] = fmaf(hv.x, w0[j], acc[j]);
            acc[j] = fmaf(hv.y, w0[OUT_DIM + j], acc[j]);
            acc[j] = fmaf(hv.z, w0[2 * OUT_DIM + j], acc[j]);
            acc[j] = fmaf(hv.w, w0[3 * OUT_DIM + j], acc[j]);
        }
    }

    float m = acc[0];
    #pragma unroll
    for (int j = 1; j < OUT_DIM; ++j) m = fmaxf(m, acc[j]);
    float s = 0.f;
    #pragma unroll
    for (int j = 0; j < OUT_DIM; ++j) s += expf(acc[j] - m);
    float lse = m + logf(s);
    float* o = out + (size_t)node * OUT_DIM;
    #pragma unroll
    for (int j = 0; j < OUT_DIM; ++j) o[j] = acc[j] - lse;
}

// ---------------------------------------------------------------------------
extern "C" void kernel_launch(void* const* d_in, const int* in_sizes, int n_in,
                              void* d_out, int out_size, void* d_ws, size_t ws_size,
                              hipStream_t stream) {
    const float*     x    = (const float*)d_in[0];
    const long long* eidx = (const long long*)d_in[1];   // int64 [2, E]
    const float*     W1   = (const float*)d_in[2];
    const float*     b1   = (const float*)d_in[3];
    const float*     W2   = (const float*)d_in[4];
    const float*     b2   = (const float*)d_in[5];
    const float*     Wl   = (const float*)d_in[6];
    const float*     bl   = (const float*)d_in[7];
    float*           out  = (float*)d_out;

    const long long* row = eidx;            // edge_index[0]
    const long long* col = eidx + N_EDGES;  // edge_index[1]

    // workspace carve-up
    char*  ws  = (char*)d_ws;
    size_t off = 0;
    float* dis = (float*)(ws + off);
    off += (((size_t)N_NODES * sizeof(float)) + 511) & ~(size_t)511;
    float* xw  = (float*)(ws + off);
    off += (((size_t)N_NODES * HID * sizeof(float)) + 511) & ~(size_t)511;
    float* agg = (float*)(ws + off);        // also reused as h1/h2

    const int TB = 256;
    const int gN   = (N_NODES + TB - 1) / TB;            // per-node
    const int gE   = (N_EDGES + TB - 1) / TB;            // per-edge
    const int gN16 = ((N_NODES * 16) + TB - 1) / TB;     // 16 thr/node
    const int gE16 = ((N_EDGES * 16) + TB - 1) / TB;     // 16 thr/edge
    const int gGemm = N_NODES / 32;                      // 100000 % 32 == 0

    // degree -> dis = rsqrt(deg)
    deg_init_kernel<<<gN, TB, 0, stream>>>(dis, N_NODES);
    deg_accum_kernel<<<gE, TB, 0, stream>>>(col, dis, N_EDGES);
    rsqrt_kernel<<<gN, TB, 0, stream>>>(dis, N_NODES);

    // ---- layer 1 ----
    gemm_wmma_kernel<IN_DIM, HID><<<gGemm, TB, 0, stream>>>(x, W1, xw);
    self_loop_kernel<<<gN16, TB, 0, stream>>>(dis, xw, agg, N_NODES);
    edge_scatter_kernel<<<gE16, TB, 0, stream>>>(row, col, dis, xw, agg, N_EDGES);
    bias_relu_kernel<<<gN16, TB, 0, stream>>>(agg, b1, N_NODES);   // agg = h1

    // ---- layer 2 ----
    gemm_wmma_kernel<HID, HID><<<gGemm, TB, 0, stream>>>(agg, W2, xw);
    self_loop_kernel<<<gN16, TB, 0, stream>>>(dis, xw, agg, N_NODES);
    edge_scatter_kernel<<<gE16, TB, 0, stream>>>(row, col, dis, xw, agg, N_EDGES);
    bias_relu_kernel<<<gN16, TB, 0, stream>>>(agg, b2, N_NODES);   // agg = h2

    // ---- linear + log_softmax ----
    logits_lsm_kernel<<<gN, TB, 0, stream>>>(agg, Wl, bl, out, N_NODES);
}